// QEqModule_42520176230553
// MI455X (gfx1250) — compile-verified
//
#include <hip/hip_runtime.h>
#include <hip/hip_bf16.h>
#include <math.h>

typedef __attribute__((ext_vector_type(2))) float v2f;
typedef __attribute__((ext_vector_type(8))) float v8f;

#define FEAT 256
#define HID 16
#define NMOL 256

// physical constants (match reference)
#define KC_EC   1.4399645478e-9f   // KC * EC
#define R0_M    2.2e-10f
#define INV_R0  (1.0f/2.2e-10f)
#define BETA_R0 8.5e10f            // BETA / R0
#define G_EW    3.0e9f
#define ANGm    1.0e-10f
#define EC_F    1.602176634e-19f

struct MlpPtrs {
  const float* W0[3]; const float* b0[3];
  const float* W1[3]; const float* b1[3];
  const float* W2[3]; const float* b2[3];
};

// ---------------------------------------------------------------------------
// K0: zero output (energies + forces) and the per-mol accumulator bins
// ---------------------------------------------------------------------------
__global__ void zero_kernel(float* __restrict__ a, long na,
                            float* __restrict__ b, long nb) {
  long i = (long)blockIdx.x * blockDim.x + threadIdx.x;
  long stride = (long)gridDim.x * blockDim.x;
  for (long j = i; j < na; j += stride) a[j] = 0.0f;
  for (long j = i; j < nb; j += stride) b[j] = 0.0f;
}

// ---------------------------------------------------------------------------
// K1: fused 3-MLP over node_feat. Layer 1 = WMMA f32 16x16x4 (fp32-exact),
// layers 2/3 per-lane through LDS. Each wave owns TWO 16-atom tiles so the
// B fragments (weights) read from LDS are shared across 32 atoms.
// A-row loads are unconditional (index clamped); only stores are guarded.
// ---------------------------------------------------------------------------
__global__ __launch_bounds__(256) void mlp3_wmma_kernel(
    const float* __restrict__ node_feat, MlpPtrs P,
    float* __restrict__ out_charge, float* __restrict__ out_en2,
    float* __restrict__ out_hard2, int n_atoms)
{
  __shared__ float sW0[3][FEAT * HID];   // 48 KB
  __shared__ float sW1[3][HID * HID];
  __shared__ float sB0[3][HID];
  __shared__ float sB1[3][HID];
  __shared__ float sW2[3][HID];
  __shared__ float sB2[3];
  __shared__ float sH1[8][16][17];       // per-wave hidden scratch, padded

  // stage weights to LDS (m unrolled -> constant pointer indices)
#pragma unroll
  for (int m = 0; m < 3; ++m) {
    for (int i = threadIdx.x; i < FEAT * HID; i += 256) sW0[m][i] = P.W0[m][i];
    for (int i = threadIdx.x; i < HID * HID; i += 256)  sW1[m][i] = P.W1[m][i];
  }
  if (threadIdx.x < 3 * HID) {
    int m = threadIdx.x / HID, r = threadIdx.x % HID;
    sB0[m][r] = P.b0[m][r];
    sB1[m][r] = P.b1[m][r];
    sW2[m][r] = P.W2[m][r];
    if (r == 0) sB2[m] = P.b2[m][0];
  }
  __syncthreads();

  const int wv   = threadIdx.x >> 5;
  const int lane = threadIdx.x & 31;
  const int nloc = lane & 15;   // M for A / N for B,D
  const int hh   = lane >> 4;   // K-half selector

  const long tile0 = ((long)blockIdx.x * 8 + wv) * 2;
  const int atom0 = (int)(tile0 * 16) + nloc;
  const int atom1 = atom0 + 16;
  // clamp row index so loads are always in-bounds (no EXEC divergence)
  const int safe0 = atom0 < n_atoms ? atom0 : (n_atoms - 1);
  const int safe1 = atom1 < n_atoms ? atom1 : (n_atoms - 1);
  const float* arow0 = node_feat + (size_t)safe0 * FEAT;
  const float* arow1 = node_feat + (size_t)safe1 * FEAT;

  v8f c0a = {}, c1a = {}, c2a = {};
  v8f c0b = {}, c1b = {}, c2b = {};

  // Layer 1: D = A(16x256) x W0(256x16), K in steps of 4 (fp32 WMMA)
  for (int k0 = 0; k0 < FEAT; k0 += 4) {
    const int ka = k0 + hh * 2;
    const float2 ta = *reinterpret_cast<const float2*>(arow0 + ka);
    const float2 tb = *reinterpret_cast<const float2*>(arow1 + ka);
    v2f a0; a0[0] = ta.x; a0[1] = ta.y;
    v2f a1; a1[0] = tb.x; a1[1] = tb.y;
    v2f b0v, b1v, b2v;
    b0v[0] = sW0[0][ka * HID + nloc];  b0v[1] = sW0[0][(ka + 1) * HID + nloc];
    b1v[0] = sW0[1][ka * HID + nloc];  b1v[1] = sW0[1][(ka + 1) * HID + nloc];
    b2v[0] = sW0[2][ka * HID + nloc];  b2v[1] = sW0[2][(ka + 1) * HID + nloc];
    c0a = __builtin_amdgcn_wmma_f32_16x16x4_f32(false, a0, false, b0v, (short)0, c0a, false, false);
    c0b = __builtin_amdgcn_wmma_f32_16x16x4_f32(false, a1, false, b0v, (short)0, c0b, false, false);
    c1a = __builtin_amdgcn_wmma_f32_16x16x4_f32(false, a0, false, b1v, (short)0, c1a, false, false);
    c1b = __builtin_amdgcn_wmma_f32_16x16x4_f32(false, a1, false, b1v, (short)0, c1b, false, false);
    c2a = __builtin_amdgcn_wmma_f32_16x16x4_f32(false, a0, false, b2v, (short)0, c2a, false, false);
    c2b = __builtin_amdgcn_wmma_f32_16x16x4_f32(false, a1, false, b2v, (short)0, c2b, false, false);
  }

  // Layers 2+3 per tile / per MLP, via LDS round-trip of h1
#pragma unroll
  for (int t = 0; t < 2; ++t) {
    const int atomT = (t == 0) ? atom0 : atom1;
    const bool validT = atomT < n_atoms;
    float res[3];
#pragma unroll
    for (int m = 0; m < 3; ++m) {
      v8f acc;
      if (t == 0) acc = (m == 0) ? c0a : ((m == 1) ? c1a : c2a);
      else        acc = (m == 0) ? c0b : ((m == 1) ? c1b : c2b);
#pragma unroll
      for (int r = 0; r < 8; ++r) {
        float v = acc[r] + sB0[m][nloc];          // D(M=r+8*hh, N=nloc)
        sH1[wv][r + 8 * hh][nloc] = v > 0.0f ? v : 0.0f;
      }
      __syncthreads();
      float rv = 0.0f;
      if (hh == 0) {                               // lanes 0..15: one atom each
        float h2[HID];
#pragma unroll
        for (int o = 0; o < HID; ++o) {
          float s = sB1[m][o];
#pragma unroll
          for (int i = 0; i < HID; ++i)
            s = fmaf(sH1[wv][nloc][i], sW1[m][i * HID + o], s);
          h2[o] = s > 0.0f ? s : 0.0f;
        }
        float s = sB2[m];
#pragma unroll
        for (int o = 0; o < HID; ++o) s = fmaf(h2[o], sW2[m][o], s);
        rv = s;
      }
      res[m] = rv;
      __syncthreads();
    }
    if (hh == 0 && validT) {
      float q = 2.0f * tanhf(res[0] * 0.5f);       // CHARGE_UB clamp
      out_charge[atomT] = q;
      out_en2[atomT]    = res[1] * res[1];
      out_hard2[atomT]  = res[2] * res[2];
    }
  }
}

// ---------------------------------------------------------------------------
// K2: per-mol sum of charge and atom count (LDS-binned, block-flush atomics)
// ---------------------------------------------------------------------------
__global__ __launch_bounds__(256) void molsum_kernel(
    const float* __restrict__ charge, const int* __restrict__ batch,
    float* __restrict__ sumq, float* __restrict__ cnt, int n)
{
  __shared__ float sq[NMOL], sc[NMOL];
  sq[threadIdx.x] = 0.0f; sc[threadIdx.x] = 0.0f;
  __syncthreads();
  long stride = (long)gridDim.x * blockDim.x;
  for (long i = (long)blockIdx.x * blockDim.x + threadIdx.x; i < n; i += stride) {
    int m = batch[i] & (NMOL - 1);
    atomicAdd(&sq[m], charge[i]);
    atomicAdd(&sc[m], 1.0f);
  }
  __syncthreads();
  atomicAdd(&sumq[threadIdx.x], sq[threadIdx.x]);
  atomicAdd(&cnt[threadIdx.x],  sc[threadIdx.x]);
}

// ---------------------------------------------------------------------------
// K3: pred_charge = charge + (total - sumq)/natoms, plus electronegativity
// self-energy accumulation per mol into out_energy
// ---------------------------------------------------------------------------
__global__ __launch_bounds__(256) void finalize_kernel(
    const float* __restrict__ charge, const float* __restrict__ en2,
    const float* __restrict__ hard2, const int* __restrict__ batch,
    const float* __restrict__ total_charge, const float* __restrict__ sumq,
    const float* __restrict__ cnt, float* __restrict__ predq,
    float* __restrict__ out_energy, int n)
{
  __shared__ float ee[NMOL];
  ee[threadIdx.x] = 0.0f;
  __syncthreads();
  long stride = (long)gridDim.x * blockDim.x;
  for (long i = (long)blockIdx.x * blockDim.x + threadIdx.x; i < n; i += stride) {
    int m = batch[i] & (NMOL - 1);
    float corr = (total_charge[m] - sumq[m]) / cnt[m];
    float q = charge[i] + corr;
    predq[i] = q;
    float e = en2[i] * q + hard2[i] * q * q * EC_F;
    atomicAdd(&ee[m], e);
  }
  __syncthreads();
  atomicAdd(&out_energy[threadIdx.x], ee[threadIdx.x]);
}

// ---------------------------------------------------------------------------
// K4: per-edge Coulomb energy/force. Energies LDS-binned per mol; forces
// scatter-added (+row / -col) with global fp32 atomics.
// ---------------------------------------------------------------------------
__global__ __launch_bounds__(256) void edge_kernel(
    const float* __restrict__ dij, const int* __restrict__ row,
    const int* __restrict__ col, const int* __restrict__ batch,
    const float* __restrict__ predq, float* __restrict__ out_energy,
    float* __restrict__ out_force, long n_edges)
{
  __shared__ float emol[NMOL];
  emol[threadIdx.x] = 0.0f;
  __syncthreads();
  long stride = (long)gridDim.x * blockDim.x;
  for (long e = (long)blockIdx.x * blockDim.x + threadIdx.x; e < n_edges; e += stride) {
    int r = row[e], c = col[e];
    size_t e3 = 3 * (size_t)e;
    float dx = dij[e3], dy = dij[e3 + 1], dz = dij[e3 + 2];
    float r2a = dx * dx + dy * dy + dz * dz;          // Angstrom^2
    float rij = sqrtf(r2a) * ANGm;                    // meters
    float inv_r = 1.0f / rij;
    float pref = KC_EC * predq[r] * predq[c] * inv_r; // eV

    float damp = 1.0f / (1.0f + expf(-(BETA_R0 * (rij - R0_M))));
    float sp   = log1pf(expf((rij - R0_M) * INV_R0));
    float ratio = (rij * INV_R0) / (1.0f + sp);

    float g = G_EW * rij;
    float expm2 = expf(-g * g);
    float t = 1.0f / (1.0f + 0.3275911f * g);
    float erfc_p = t * (0.254829592f + t * (-0.284496736f +
                   t * (1.421413741f + t * (-1.453152027f +
                   t * 1.061405429f)))) * expm2;

    float ecoul = pref * ratio + pref * (erfc_p - 1.0f);
    float fcoul = pref * damp * ratio * ratio
                + pref * (erfc_p + 2.0f * g * expm2 - 1.0f);

    float fs = fcoul * ANGm * inv_r * inv_r;          // multiplies dij[Ang]
    float fx = dx * fs, fy = dy * fs, fz = dz * fs;

    atomicAdd(&emol[batch[r] & (NMOL - 1)], 0.5f * ecoul);
    atomicAdd(&out_force[3 * (size_t)r + 0],  fx);
    atomicAdd(&out_force[3 * (size_t)r + 1],  fy);
    atomicAdd(&out_force[3 * (size_t)r + 2],  fz);
    atomicAdd(&out_force[3 * (size_t)c + 0], -fx);
    atomicAdd(&out_force[3 * (size_t)c + 1], -fy);
    atomicAdd(&out_force[3 * (size_t)c + 2], -fz);
  }
  __syncthreads();
  atomicAdd(&out_energy[threadIdx.x], emol[threadIdx.x]);
}

// ---------------------------------------------------------------------------
extern "C" void kernel_launch(void* const* d_in, const int* in_sizes, int n_in,
                              void* d_out, int out_size, void* d_ws, size_t ws_size,
                              hipStream_t stream) {
  const float* node_feat    = (const float*)d_in[0];
  const float* dij          = (const float*)d_in[1];
  const float* total_charge = (const float*)d_in[2];
  const int*   row          = (const int*)d_in[3];
  const int*   col          = (const int*)d_in[4];
  const int*   batch        = (const int*)d_in[5];

  MlpPtrs P;
  for (int m = 0; m < 3; ++m) {
    int base = 6 + 6 * m;
    P.W0[m] = (const float*)d_in[base + 0];
    P.b0[m] = (const float*)d_in[base + 1];
    P.W1[m] = (const float*)d_in[base + 2];
    P.b1[m] = (const float*)d_in[base + 3];
    P.W2[m] = (const float*)d_in[base + 4];
    P.b2[m] = (const float*)d_in[base + 5];
  }

  const int  n_atoms = in_sizes[0] / FEAT;   // 200000
  const long n_edges = in_sizes[3];          // 6400000
  const int  n_mols  = in_sizes[2];          // 256

  float* outp       = (float*)d_out;
  float* out_energy = outp;                  // [256]
  float* out_force  = outp + n_mols;         // [n_atoms*3]

  float* ws        = (float*)d_ws;
  float* ws_charge = ws;
  float* ws_en2    = ws + (size_t)n_atoms;
  float* ws_hard2  = ws + 2 * (size_t)n_atoms;
  float* ws_predq  = ws + 3 * (size_t)n_atoms;
  float* ws_sumq   = ws + 4 * (size_t)n_atoms;
  float* ws_cnt    = ws_sumq + NMOL;

  // K0: zero energies+forces and mol bins
  long nz = (long)n_mols + 3L * n_atoms;
  int zb = (int)((nz + 255) / 256);
  zero_kernel<<<zb, 256, 0, stream>>>(outp, nz, ws_sumq, 2 * NMOL);

  // K1: fused MLPs (WMMA), 2 tiles per wave -> 32 atoms/wave, 256 atoms/block
  int tiles  = (n_atoms + 15) / 16;
  int blocks = (tiles + 15) / 16;
  mlp3_wmma_kernel<<<blocks, 256, 0, stream>>>(node_feat, P, ws_charge,
                                               ws_en2, ws_hard2, n_atoms);

  // K2: per-mol charge sums / counts
  molsum_kernel<<<512, 256, 0, stream>>>(ws_charge, batch, ws_sumq, ws_cnt, n_atoms);

  // K3: pred_charge + electronegativity energy
  finalize_kernel<<<512, 256, 0, stream>>>(ws_charge, ws_en2, ws_hard2, batch,
                                           total_charge, ws_sumq, ws_cnt,
                                           ws_predq, out_energy, n_atoms);

  // K4: edge Coulomb energy + forces
  edge_kernel<<<2048, 256, 0, stream>>>(dij, row, col, batch, ws_predq,
                                        out_energy, out_force, n_edges);
}